// MeshPool_40896678592649
// MI455X (gfx1250) — compile-verified
//
#include <hip/hip_runtime.h>
#include <hip/hip_bf16.h>
#include <stdint.h>

#define NT 1024
#define MAXN 6144

typedef __attribute__((ext_vector_type(2))) float v2f;
typedef __attribute__((ext_vector_type(8))) float v8f;

// ---------------- bitonic argsort (descending value, ascending index on ties) ----------------
__global__ __launch_bounds__(NT) void k_sort(const float* __restrict__ pool,
                                             int* __restrict__ order, int n) {
  const int M = 8192;
  __shared__ unsigned long long keys[M];
  int tid = threadIdx.x;
  for (int i = tid; i < M; i += NT) {
    unsigned long long kk = ~0ull;
    if (i < n) {
      unsigned int b = __float_as_uint(pool[i]);
      unsigned int asc = b ^ ((unsigned int)((int)b >> 31) | 0x80000000u); // order-preserving map
      unsigned int desc = ~asc;                                            // descending
      kk = ((unsigned long long)desc << 32) | (unsigned int)i;             // stable ties by index
    }
    keys[i] = kk;
  }
  __syncthreads();
  for (int k = 2; k <= M; k <<= 1) {
    for (int j = k >> 1; j > 0; j >>= 1) {
      for (int i = tid; i < M; i += NT) {
        int ixj = i ^ j;
        if (ixj > i) {
          bool up = ((i & k) == 0);
          unsigned long long a = keys[i], b = keys[ixj];
          if ((a > b) == up) { keys[i] = b; keys[ixj] = a; }
        }
      }
      __syncthreads();
    }
  }
  for (int i = tid; i < n; i += NT) order[i] = (int)(unsigned int)keys[i];
}

// ---------------- init ----------------
__global__ void k_init(int* __restrict__ clusW, float* __restrict__ scaleW,
                       int* __restrict__ counts, int n) {
  int i = blockIdx.x * blockDim.x + threadIdx.x;
  if (i < n) { clusW[i] = 0; scaleW[i] = 1.0f; counts[i] = 0; }
}

// async global -> LDS row copy, b64 per lane, uniform 3 issues per wave for n=6144
__device__ __forceinline__ void row_async(const float* __restrict__ src,
                                          float* __restrict__ dstLds, int n, int tid) {
  for (int p = tid; p < (n >> 1); p += NT) {
    unsigned int laddr = (unsigned int)(unsigned long long)&dstLds[2 * p];
    const float* gp = src + 2 * p;
    asm volatile("global_load_async_to_lds_b64 %0, %1, off"
                 :: "v"(laddr), "v"(gp) : "memory");
  }
}

// ---------------- serial greedy merge (one workgroup, async double-buffered rows) ----------------
__global__ __launch_bounds__(NT) void k_greedy(
    const float* __restrict__ adj, const int* __restrict__ order,
    int* __restrict__ clusW, float* __restrict__ scaleW,
    int* __restrict__ nclusW, int* __restrict__ outClus,
    int n, int target) {
  __shared__ __align__(16) float rowbuf[2][MAXN];
  __shared__ unsigned char rem[MAXN];
  __shared__ int part[NT];       // 32 used per step; full array used by epilogue scan
  __shared__ float partf[32];
  __shared__ int s_i, s_remTot, s_stop, s_flags, s_iUsed;
  __shared__ float s_dsafe;

  int tid = threadIdx.x;
  int lane = tid & 31;
  int wid = tid >> 5;
  for (int j = tid; j < n; j += NT) rem[j] = 1;
  if (tid == 0) { s_i = 0; s_remTot = n; s_stop = 0; }
  __syncthreads();

  // prologue: prefetch row order[0] into buffer 0
  row_async(adj + (size_t)order[0] * n, rowbuf[0], n, tid);

  for (int t = 0; t < n; ++t) {
    int idx = order[t];
    if (t + 1 < n) {
      row_async(adj + (size_t)order[t + 1] * n, rowbuf[(t + 1) & 1], n, tid);
      asm volatile("s_wait_asynccnt 3" ::: "memory"); // current row (3 older ops) complete
    } else {
      asm volatile("s_wait_asynccnt 0" ::: "memory");
    }
    __syncthreads();

    const float* row = rowbuf[t & 1];
    int cloc = 0; float dloc = 0.0f;
    for (int j = tid; j < n; j += NT) {
      float w = row[j];
      if (w > 0.0f && rem[j]) { cloc++; dloc += w; }
    }
    // wave-level reduction (wave32, deterministic fixed tree, no barriers)
#pragma unroll
    for (int m = 16; m > 0; m >>= 1) {
      cloc += __shfl_down(cloc, m, 32);
      dloc += __shfl_down(dloc, m, 32);
    }
    if (lane == 0) { part[wid] = cloc; partf[wid] = dloc; }
    __syncthreads();
    if (tid < 32) {
      int c2 = part[tid];
      float d2 = partf[tid];
#pragma unroll
      for (int m = 16; m > 0; m >>= 1) {
        c2 += __shfl_down(c2, m, 32);
        d2 += __shfl_down(d2, m, 32);
      }
      if (tid == 0) {
        int cnt = c2; float den = d2;
        int active = (!s_stop) && rem[idx];
        int single = (cnt == 1);
        int dogrp = active && !single;
        s_iUsed = s_i;
        s_dsafe = (den > 0.0f) ? den : 1.0f;
        s_flags = (active ? 1 : 0) | (dogrp ? 2 : 0);
        if (active) {
          s_remTot -= cnt;
          if (single) clusW[idx] = s_i;
          s_i += 1;
          if (dogrp && s_remTot <= target) s_stop = 1;
        }
      }
    }
    __syncthreads();
    if (s_flags & 1) {
      int iu = s_iUsed;
      float ds = s_dsafe;
      int grp = s_flags & 2;
      for (int j = tid; j < n; j += NT) {
        float w = row[j];
        if (w > 0.0f && rem[j]) {
          rem[j] = 0;
          if (grp) { clusW[j] = iu; scaleW[j] *= w / ds; }
        }
      }
    }
    __syncthreads();
    if (s_stop) break;
  }
  __syncthreads();

  // leftover vertices -> fresh clusters in ascending index order (block prefix sum)
  const int CH = (n + NT - 1) / NT;
  int base = tid * CH;
  int sloc[8];
  int run = 0;
  for (int k = 0; k < CH && k < 8; ++k) {
    int v = base + k;
    sloc[k] = run;
    if (v < n && rem[v]) run++;
  }
  part[tid] = run;
  __syncthreads();
  for (int off = 1; off < NT; off <<= 1) {
    int add = (tid >= off) ? part[tid - off] : 0;
    __syncthreads();
    part[tid] += add;
    __syncthreads();
  }
  int excl = tid ? part[tid - 1] : 0;
  int ibase = s_i;
  for (int k = 0; k < CH && k < 8; ++k) {
    int v = base + k;
    if (v < n) {
      int c = rem[v] ? (ibase + excl + sloc[k]) : clusW[v];
      clusW[v] = c;
      outClus[v] = c;
    }
  }
  if (tid == NT - 1) *nclusW = ibase + part[NT - 1];
}

// ---------------- counting sort by cluster ----------------
__global__ void k_hist(const int* __restrict__ clus, int* __restrict__ counts, int n) {
  int v = blockIdx.x * blockDim.x + threadIdx.x;
  if (v < n) atomicAdd(&counts[clus[v]], 1);      // int atomics: deterministic result
}

__global__ __launch_bounds__(NT) void k_scan(const int* __restrict__ counts,
                                             int* __restrict__ starts, int n) {
  __shared__ int part[NT];
  int tid = threadIdx.x;
  const int CH = (n + NT - 1) / NT;
  int base = tid * CH;
  int loc[8]; int s = 0;
  for (int k = 0; k < CH && k < 8; ++k) {
    int v = base + k;
    loc[k] = s;
    if (v < n) s += counts[v];
  }
  part[tid] = s;
  __syncthreads();
  for (int off = 1; off < NT; off <<= 1) {
    int add = (tid >= off) ? part[tid - off] : 0;
    __syncthreads();
    part[tid] += add;
    __syncthreads();
  }
  int excl = tid ? part[tid - 1] : 0;
  for (int k = 0; k < CH && k < 8; ++k) {
    int v = base + k;
    if (v < n) starts[v] = excl + loc[k];
  }
  if (tid == NT - 1) starts[n] = part[NT - 1];
}

// deterministic stable scatter: rank = #{u < v : cluster[u]==cluster[v]}
__global__ void k_rankscatter(const int* __restrict__ clus, const int* __restrict__ starts,
                              int* __restrict__ sortedV, int n) {
  int v = blockIdx.x * blockDim.x + threadIdx.x;
  if (v >= n) return;
  int c = clus[v];
  int r = 0;
  for (int u = 0; u < v; ++u) r += (clus[u] == c) ? 1 : 0;
  sortedV[starts[c] + r] = v;
}

// ---------------- segment-sum via V_WMMA_F32_16X16X4_F32 ----------------
// One wave per (16-cluster tile, 16-feature tile). A = one-hot*scale selection, B = features.
// Vertex indices / clusters / scales for each 4-chunk are wave-uniform -> scalar loads,
// only the 16-wide feature row load stays per-lane (coalesced).
__global__ __launch_bounds__(32) void k_segsum(
    const float* __restrict__ x, const float* __restrict__ verts,
    const int* __restrict__ clus, const float* __restrict__ scale,
    const int* __restrict__ sortedV, const int* __restrict__ starts,
    float* __restrict__ out_x, float* __restrict__ out_v,
    int n, int d, int ftiles) {
  int lane = threadIdx.x;
  int half = lane >> 4;     // lanes 0-15 vs 16-31
  int ln = lane & 15;
  int c0 = blockIdx.x * 16;
  int yt = blockIdx.y;      // 0..ftiles-1: features, ==ftiles: vertices(3)
  int rbeg = starts[c0];
  int rend = starts[c0 + 16];
  v8f acc = {};
  for (int base = rbeg; base < rend; base += 4) {
    // uniform scalar fetches for the 4 vertices of this chunk
    int v0 = sortedV[base];
    int v1 = (base + 1 < rend) ? sortedV[base + 1] : -1;
    int v2 = (base + 2 < rend) ? sortedV[base + 2] : -1;
    int v3 = (base + 3 < rend) ? sortedV[base + 3] : -1;
    int  cc0 = clus[v0];               float ss0 = scale[v0];
    int  cc1 = (v1 >= 0) ? clus[v1] : -1; float ss1 = (v1 >= 0) ? scale[v1] : 0.0f;
    int  cc2 = (v2 >= 0) ? clus[v2] : -1; float ss2 = (v2 >= 0) ? scale[v2] : 0.0f;
    int  cc3 = (v3 >= 0) ? clus[v3] : -1; float ss3 = (v3 >= 0) ? scale[v3] : 0.0f;
    // per half-wave K selection: half 0 -> K={0,1}, half 1 -> K={2,3}
    int   vlo = half ? v2  : v0;  int   clo = half ? cc2 : cc0;  float slo = half ? ss2 : ss0;
    int   vhi = half ? v3  : v1;  int   chi = half ? cc3 : cc1;  float shi = half ? ss3 : ss1;
    v2f a, b;
    a[0] = (clo == c0 + ln) ? slo : 0.0f;   // A[m=ln][k_lo]
    a[1] = (chi == c0 + ln) ? shi : 0.0f;   // A[m=ln][k_hi]
    float blo = 0.0f, bhi = 0.0f;
    if (yt < ftiles) {
      int col = 16 * yt + ln;
      if (vlo >= 0) blo = x[(size_t)vlo * d + col];
      if (vhi >= 0) bhi = x[(size_t)vhi * d + col];
    } else {
      if (vlo >= 0 && ln < 3) blo = verts[vlo * 3 + ln];
      if (vhi >= 0 && ln < 3) bhi = verts[vhi * 3 + ln];
    }
    b[0] = blo;                              // B[k_lo][n=ln]
    b[1] = bhi;                              // B[k_hi][n=ln]
    acc = __builtin_amdgcn_wmma_f32_16x16x4_f32(false, a, false, b, (short)0, acc,
                                                false, false);
  }
#pragma unroll
  for (int i = 0; i < 8; ++i) {
    int m = i + 8 * half;                    // C/D: VGPR i -> M=i (lanes 0-15) / M=i+8 (16-31)
    int row = c0 + m;
    if (yt < ftiles) out_x[(size_t)row * d + 16 * yt + ln] = acc[i];
    else if (ln < 3) out_v[row * 3 + ln] = acc[i];
  }
}

// ---------------- edge / face remap ----------------
__global__ void k_edges(const int* __restrict__ edges, const int* __restrict__ clus,
                        const int* __restrict__ nclusP, int* __restrict__ out_e,
                        float* __restrict__ out_m, int n, int e) {
  int i = blockIdx.x * blockDim.x + threadIdx.x;
  int T = n + e;
  if (i >= T) return;
  if (i < n) {
    out_e[i] = i; out_e[T + i] = i;
    out_m[i] = (i < *nclusP) ? 1.0f : 0.0f;
  } else {
    int k = i - n;
    int c0 = clus[edges[k]];
    int c1 = clus[edges[e + k]];
    out_e[i] = c0; out_e[T + i] = c1;
    out_m[i] = (c0 != c1) ? 1.0f : 0.0f;
  }
}

__global__ void k_faces(const int* __restrict__ faces, const int* __restrict__ clus,
                        int* __restrict__ out_f, float* __restrict__ out_m, int f) {
  int i = blockIdx.x * blockDim.x + threadIdx.x;
  if (i >= f) return;
  int a = clus[faces[i]];
  int b = clus[faces[f + i]];
  int c = clus[faces[2 * f + i]];
  out_f[i * 3 + 0] = a;
  out_f[i * 3 + 1] = b;
  out_f[i * 3 + 2] = c;
  out_m[i] = (a != b && a != c && b != c) ? 1.0f : 0.0f;
}

extern "C" void kernel_launch(void* const* d_in, const int* in_sizes, int n_in,
                              void* d_out, int out_size, void* d_ws, size_t ws_size,
                              hipStream_t stream) {
  const float* x     = (const float*)d_in[0];
  const float* verts = (const float*)d_in[1];
  const int*   edges = (const int*)d_in[2];
  const int*   faces = (const int*)d_in[3];
  const float* pool  = (const float*)d_in[4];
  const float* adj   = (const float*)d_in[5];

  int N = in_sizes[4];
  int D = in_sizes[0] / N;
  int E = in_sizes[2] / 2;
  int F = in_sizes[3] / 3;

  // output layout (reference return order, flat)
  float* out_x  = (float*)d_out;                          // N*D
  float* out_v  = out_x + (size_t)N * D;                  // N*3
  int*   out_e  = (int*)(out_v + (size_t)N * 3);          // 2*(N+E)
  float* out_em = (float*)(out_e + (size_t)2 * (N + E));  // N+E
  int*   out_f  = (int*)(out_em + (size_t)(N + E));       // F*3
  float* out_fm = (float*)(out_f + (size_t)F * 3);        // F
  int*   out_c  = (int*)(out_fm + (size_t)F);             // N

  // workspace layout
  int*   order   = (int*)d_ws;          // N
  int*   clusW   = order + N;           // N
  float* scaleW  = (float*)(clusW + N); // N
  int*   counts  = (int*)(scaleW + N);  // N
  int*   startsW = counts + N;          // N+1
  int*   sortedW = startsW + N + 1;     // N
  int*   nclusW  = sortedW + N;         // 1

  int nb = (N + 255) / 256;
  k_sort<<<1, NT, 0, stream>>>(pool, order, N);
  k_init<<<nb, 256, 0, stream>>>(clusW, scaleW, counts, N);
  k_greedy<<<1, NT, 0, stream>>>(adj, order, clusW, scaleW, nclusW, out_c, N, N / 2);
  k_hist<<<nb, 256, 0, stream>>>(clusW, counts, N);
  k_scan<<<1, NT, 0, stream>>>(counts, startsW, N);
  k_rankscatter<<<nb, 256, 0, stream>>>(clusW, startsW, sortedW, N);

  int ftiles = D / 16;
  dim3 g7(N / 16, ftiles + 1);
  k_segsum<<<g7, 32, 0, stream>>>(x, verts, clusW, scaleW, sortedW, startsW,
                                  out_x, out_v, N, D, ftiles);

  int tot = N + E;
  k_edges<<<(tot + 255) / 256, 256, 0, stream>>>(edges, clusW, nclusW, out_e, out_em, N, E);
  k_faces<<<(F + 255) / 256, 256, 0, stream>>>(faces, clusW, out_f, out_fm, F);
}